// EntropyMetircs_2d_83288005804273
// MI455X (gfx1250) — compile-verified
//
#include <hip/hip_runtime.h>
#include <hip/hip_bf16.h>

#define HH 1024
#define WW 1024
#define NBINS_MEAN 512
#define HIST_BINS (256 * NBINS_MEAN)   // 131072 bins per image
#define TILE 32
#define HALO (TILE + 2)
#define WS_HIST_OFF 64                 // uints (256 B) reserved for max slot

typedef __attribute__((ext_vector_type(2))) float v2f;
typedef __attribute__((ext_vector_type(8))) float v8f;

#if __has_builtin(__builtin_amdgcn_global_load_async_to_lds_b32)
#define HAVE_ASYNC_LDS 1
#else
#define HAVE_ASYNC_LDS 0
#endif

// builtin prototype (from hipcc diagnostic): global/LDS *typed* pointers
#define AS1I(p) ((__attribute__((address_space(1))) int*)(p))
#define AS3I(p) ((__attribute__((address_space(3))) int*)(p))

// ---------------- init: zero max slot, histograms, and output -----------------
__global__ void init_kernel(unsigned int* __restrict__ ws, float* __restrict__ out,
                            long long n_ws_uints) {
    long long i = (long long)blockIdx.x * blockDim.x + threadIdx.x;
    long long stride = (long long)gridDim.x * blockDim.x;
    for (; i < n_ws_uints; i += stride) ws[i] = 0u;
    if (blockIdx.x == 0 && threadIdx.x == 0) out[0] = 0.0f;
}

// ---------------- global max of input (x >= 0, so int-bit compare is monotonic)
__global__ void __launch_bounds__(256) max_kernel(const float4* __restrict__ x4,
                                                  unsigned int* __restrict__ maxbits,
                                                  long long n4) {
    float m = 0.0f;
    long long i = (long long)blockIdx.x * blockDim.x + threadIdx.x;
    long long stride = (long long)gridDim.x * blockDim.x;
    for (; i < n4; i += stride) {
        float4 v = x4[i];
        m = fmaxf(m, fmaxf(fmaxf(v.x, v.y), fmaxf(v.z, v.w)));
    }
    __shared__ float red[256];
    red[threadIdx.x] = m;
    __syncthreads();
    for (int s = 128; s > 0; s >>= 1) {
        if (threadIdx.x < s) red[threadIdx.x] = fmaxf(red[threadIdx.x], red[threadIdx.x + s]);
        __syncthreads();
    }
    if (threadIdx.x == 0) atomicMax((int*)maxbits, __float_as_int(red[0]));
}

// ---------------- quantize + 8-neighbor sum + joint histogram -----------------
// One 32x32 output tile per block; 34x34 halo tile staged through LDS.
// Interior blocks use ASYNCcnt-tracked global->LDS async copies (CDNA5 path);
// border blocks take a guarded synchronous path (branch is uniform per block).
// Histograms live in global memory and stay resident in the 192 MB L2.
__global__ void __launch_bounds__(256) hist_kernel(const float* __restrict__ x,
                                                   const unsigned int* __restrict__ maxbits,
                                                   unsigned int* __restrict__ hist,
                                                   int B) {
    __shared__ float ftile[HALO * HALO];
    __shared__ int   tile[HALO * HALO];
    const int tid = threadIdx.x;
    const int b  = blockIdx.z;
    const int ox = blockIdx.x * TILE;
    const int oy = blockIdx.y * TILE;

    const float scale = (__uint_as_float(maxbits[0]) < 1.0f) ? 255.0f : 1.0f;
    const float* img = x + (long long)b * HH * WW;

    const bool interior = (ox > 0) && (oy > 0) && (ox + TILE < WW) && (oy + TILE < HH);
    if (interior) {
#if HAVE_ASYNC_LDS
        for (int i = tid; i < HALO * HALO; i += 256) {
            int r = i / HALO, c = i - r * HALO;
            const float* g = img + (long long)(oy - 1 + r) * WW + (ox - 1 + c);
            __builtin_amdgcn_global_load_async_to_lds_b32(AS1I(g), AS3I(&ftile[i]),
                                                          /*offset=*/0, /*cpol=*/0);
        }
        asm volatile("s_wait_asynccnt 0" ::: "memory");
#else
        for (int i = tid; i < HALO * HALO; i += 256) {
            int r = i / HALO, c = i - r * HALO;
            ftile[i] = img[(long long)(oy - 1 + r) * WW + (ox - 1 + c)];
        }
#endif
    } else {
        for (int i = tid; i < HALO * HALO; i += 256) {
            int r = i / HALO, c = i - r * HALO;
            int gr = oy + r - 1, gc = ox + c - 1;
            float v = 0.0f;
            if (gr >= 0 && gr < HH && gc >= 0 && gc < WW)
                v = img[(long long)gr * WW + gc];
            ftile[i] = v;
        }
    }
    __syncthreads();

    // quantize the staged tile: truncating uint8 cast, matches astype(uint8)
    for (int i = tid; i < HALO * HALO; i += 256)
        tile[i] = (int)(unsigned char)(ftile[i] * scale);
    __syncthreads();

    unsigned int* hb = hist + (long long)b * HIST_BINS;
    for (int p = tid; p < TILE * TILE; p += 256) {
        int lr = p >> 5, lc = p & 31;
        const int* t = &tile[(lr + 1) * HALO + (lc + 1)];
        int nb = t[-HALO - 1] + t[-HALO] + t[-HALO + 1]
               + t[-1]                   + t[1]
               + t[ HALO - 1] + t[ HALO] + t[ HALO + 1];
        int row = oy + lr;
        // divisor depends on (batch,row): 3 at the four (b,r) corners, else 5
        int div = ((row == 0 || row == HH - 1) && (b == 0 || b == B - 1)) ? 3 : 5;
        int mean_i = nb / div;                 // nb >= 0: trunc == floor
        int code = t[0] * NBINS_MEAN + mean_i;
        atomicAdd(&hb[code], 1u);              // L2-resident histogram atomic
    }
}

// ---------------- entropy per image, WMMA-based wave reduction ----------------
__global__ void __launch_bounds__(256) entropy_kernel(const unsigned int* __restrict__ hist,
                                                      float* __restrict__ out, float invB) {
    const int b = blockIdx.x;
    const unsigned int* hb = hist + (long long)b * HIST_BINS;
    const float invN = 1.0f / (float)(HH * WW);

    float s = 0.0f;
    for (int i = threadIdx.x; i < HIST_BINS; i += 256) {   // 512 uniform iterations
        unsigned int c = hb[i];
        if (c) {
            float p = (float)c * invN;
            s -= p * __log2f(p);
        }
    }

    // Wave32 sum via V_WMMA_F32_16X16X4_F32 against an all-ones B matrix.
    // A layout (16x4 f32): lane L<16 -> A[L,0]=a.x, A[L,1]=a.y;
    //                      lane L>=16 -> A[L-16,2]=a.x, A[L-16,3]=a.y.
    // With a = {s, 0}: D[m,n] = s_m + s_{m+16} for every column n.
    // EXEC is all ones here: full 256-thread block, no divergence at this point.
    v2f a;  a.x = s;    a.y = 0.0f;
    v2f bb; bb.x = 1.0f; bb.y = 1.0f;
    v8f c8 = {};
    c8 = __builtin_amdgcn_wmma_f32_16x16x4_f32(false, a, false, bb,
                                               (short)0, c8, false, false);
    // Each lane holds 8 rows of its column; lanes n and n+16 cover disjoint row halves.
    float t = c8[0] + c8[1] + c8[2] + c8[3] + c8[4] + c8[5] + c8[6] + c8[7];
    t += __shfl_xor(t, 16, 32);            // combine the two row halves -> full wave sum

    if ((threadIdx.x & 31) == 0)
        atomicAdd(out, t * invB);          // 8 waves/block x 64 blocks = 512 atomics
}

extern "C" void kernel_launch(void* const* d_in, const int* in_sizes, int n_in,
                              void* d_out, int out_size, void* d_ws, size_t ws_size,
                              hipStream_t stream) {
    const float* x = (const float*)d_in[0];
    const long long n = (long long)in_sizes[0];
    const int B = (int)(n / (long long)(HH * WW));     // 64 for the reference shapes

    unsigned int* ws      = (unsigned int*)d_ws;
    unsigned int* maxbits = ws;                        // [0]
    unsigned int* hist    = ws + WS_HIST_OFF;          // 32 MB of per-image histograms
    float* out = (float*)d_out;

    const long long n_ws_uints = WS_HIST_OFF + (long long)B * HIST_BINS;

    init_kernel<<<2048, 256, 0, stream>>>(ws, out, n_ws_uints);
    max_kernel<<<1024, 256, 0, stream>>>((const float4*)x, maxbits, n / 4);

    dim3 hgrid(WW / TILE, HH / TILE, B);               // (32, 32, 64)
    hist_kernel<<<hgrid, 256, 0, stream>>>(x, maxbits, hist, B);

    entropy_kernel<<<B, 256, 0, stream>>>(hist, out, 1.0f / (float)B);
}